// SIGRN_9620726743076
// MI455X (gfx1250) — compile-verified
//
#include <hip/hip_runtime.h>
#include <math.h>

// ---------------------------------------------------------------------------
// SIGRN forward pass for MI455X (gfx1250), wave32, f32 WMMA (16x16x4).
//   B=1024, N_GENE=1000, HIDDEN=128, Z_DIM=1
// Pipeline:
//   pre -> fused inf-MLP (WMMA) -> zp@IA GEMM x2 (WMMA) -> z/KL ->
//   build IA -> in-place Gauss-Jordan inverse (persistent, grid barrier) ->
//   z@IAinv GEMM (WMMA) -> fused gen-MLP (WMMA) -> losses
// ---------------------------------------------------------------------------

#define BB    1024
#define NG    1000
#define BN    (BB * NG)          // 1,024,000 elements
#define HID   128
#define PITCH 132                // LDS row pitch (floats) to dodge bank conflicts

typedef __attribute__((ext_vector_type(2))) float v2f;
typedef __attribute__((ext_vector_type(8))) float v8f;

// ---------------------------------------------------------------------------
// Kernel 1: xn = where(da_rand < 0.1, 0, x); (xn - mean)/std; non-finite -> 0
// ---------------------------------------------------------------------------
__global__ void sigrn_pre(const float* __restrict__ x,
                          const float* __restrict__ gmean,
                          const float* __restrict__ gstd,
                          const float* __restrict__ da,
                          float* __restrict__ xn) {
    int i = blockIdx.x * 256 + threadIdx.x;     // grid covers BN exactly
    float v = (da[i] < 0.1f) ? 0.0f : x[i];
    float r = (v - gmean[0]) / gstd[0];
    if (!isfinite(r)) r = 0.0f;
    xn[i] = r;
}

// ---------------------------------------------------------------------------
// Kernel 2: fused per-element MLP.  Each wave handles 16 elements.
//   h1 = tanh(in*w1 + b1)  (LDS)
//   h2 = tanh(H1 @ W2^T + b2)   -- 8 N-tiles of V_WMMA_F32_16X16X4_F32
//   out_d = W3 h2 + b3          -- shfl_xor reduction from C-layout regs
// blockDim = 128 (4 waves) -> 64 elements per block.
// ---------------------------------------------------------------------------
__global__ void sigrn_mlp(const float* __restrict__ in,
                          const float* __restrict__ w1, const float* __restrict__ b1,
                          const float* __restrict__ w2, const float* __restrict__ b2,
                          const float* __restrict__ w3, const float* __restrict__ b3,
                          int outDim,
                          float* __restrict__ out0, float* __restrict__ out1) {
    __shared__ float H1[4][16 * PITCH];

    const int tid  = threadIdx.x;
    const int wave = tid >> 5;
    const int lane = tid & 31;
    const int hlf  = lane >> 4;      // 0: lanes 0-15, 1: lanes 16-31
    const int le   = lane & 15;      // element-in-tile / column index
    const int base = blockIdx.x * 64 + wave * 16;

    // ---- layer 1: each lane fills half the K range of its element's h1 row
    const float xv = in[base + le];
    #pragma unroll 4
    for (int kk = 0; kk < 64; ++kk) {
        int k = hlf * 64 + kk;
        H1[wave][le * PITCH + k] = tanhf(fmaf(xv, w1[k], b1[k]));
    }
    __syncthreads();   // (per-wave buffers; barrier is belt-and-braces)

    // ---- layer 2: 8 output tiles of 16 columns, K=128 in steps of 4
    const int koff = hlf * 2;        // A/B fragment K offset per lane half
    v8f h2t[8];
    for (int nt = 0; nt < 8; ++nt) {
        const int n = nt * 16 + le;          // hidden unit (B column)
        const float bb = b2[n];
        v8f c;
        #pragma unroll
        for (int j = 0; j < 8; ++j) c[j] = bb;

        #pragma unroll 8
        for (int kc = 0; kc < 32; ++kc) {
            const int k0 = kc * 4 + koff;
            v2f a, b;
            a.x = H1[wave][le * PITCH + k0];
            a.y = H1[wave][le * PITCH + k0 + 1];
            b.x = w2[n * HID + k0];          // B[k][n] = W2[n][k]
            b.y = w2[n * HID + k0 + 1];
            c = __builtin_amdgcn_wmma_f32_16x16x4_f32(
                    false, a, false, b, (short)0, c, false, false);
        }
        v8f t;
        #pragma unroll
        for (int j = 0; j < 8; ++j) t[j] = tanhf(c[j]);
        h2t[nt] = t;
    }

    // ---- layer 3: out_d[e] = b3[d] + sum_n w3[d][n]*h2[e][n]
    // C layout: register j holds row M = j + 8*hlf, column N = nt*16 + le.
    #pragma unroll
    for (int j = 0; j < 8; ++j) {
        float s0 = 0.0f, s1 = 0.0f;
        #pragma unroll
        for (int nt = 0; nt < 8; ++nt) {
            const float v = h2t[nt][j];
            const int n = nt * 16 + le;
            s0 = fmaf(w3[n], v, s0);
            if (outDim > 1) s1 = fmaf(w3[HID + n], v, s1);
        }
        for (int m = 1; m <= 8; m <<= 1) {
            s0 += __shfl_xor(s0, m, 32);
            if (outDim > 1) s1 += __shfl_xor(s1, m, 32);
        }
        if (le == 0) {
            const int e = base + j + hlf * 8;
            out0[e] = s0 + b3[0];
            if (outDim > 1) out1[e] = s1 + b3[1];
        }
    }
}

// ---------------------------------------------------------------------------
// Kernel 3: C[o,h] = sum_g A[o,g] * B[g,h]   (M=1024, N=1000, K=1000)
//   mode 0: B[g,h] = (g==h) ? 1 : -adjA[g,h]   (IA built on the fly)
//   mode 1: B[g,h] = Bsrc[g,h]                 (IA inverse)
// One 16x16 tile per wave; 64 x 63 tiles; blockDim=128 -> 1008 blocks.
// ---------------------------------------------------------------------------
__global__ void sigrn_gemm(const float* __restrict__ A,
                           const float* __restrict__ Bsrc,
                           float* __restrict__ out, int mode) {
    const int tid  = threadIdx.x;
    const int wave = tid >> 5;
    const int lane = tid & 31;
    const int hlf  = lane >> 4;
    const int le   = lane & 15;

    const int t  = blockIdx.x * 4 + wave;   // 0..4031
    const int mt = t / 63, nt = t % 63;
    const int obase = mt * 16, hbase = nt * 16;

    const int  h   = hbase + le;
    const bool hok = (h < NG);
    const int  hc  = hok ? h : (NG - 1);    // clamped, branch-free loads

    v8f c;
    #pragma unroll
    for (int j = 0; j < 8; ++j) c[j] = 0.0f;

    const int koff = hlf * 2;
    const float* Arow = A + (obase + le) * NG;

    for (int g0 = 0; g0 < NG; g0 += 4) {
        const int ga = g0 + koff;
        v2f a, b;
        a.x = Arow[ga];
        a.y = Arow[ga + 1];
        float v0 = Bsrc[ga * NG + hc];
        float v1 = Bsrc[(ga + 1) * NG + hc];
        if (!hok) { v0 = 0.0f; v1 = 0.0f; }
        if (mode == 0) {
            b.x = (ga     == h) ? 1.0f : -v0;
            b.y = (ga + 1 == h) ? 1.0f : -v1;
        } else {
            b.x = v0;
            b.y = v1;
        }
        c = __builtin_amdgcn_wmma_f32_16x16x4_f32(
                false, a, false, b, (short)0, c, false, false);
    }

    if (hok) {
        #pragma unroll
        for (int j = 0; j < 8; ++j) {
            const int row = obase + j + hlf * 8;
            out[row * NG + h] = c[j];
        }
    }
}

// ---------------------------------------------------------------------------
// Kernel 4: z = mu + exp(0.5*logvar)*eps ; per-block KL partial sums
// ---------------------------------------------------------------------------
__global__ void sigrn_z_kl(const float* __restrict__ mu,
                           const float* __restrict__ lv,
                           const float* __restrict__ eps,
                           float* __restrict__ z,
                           float* __restrict__ part) {
    __shared__ float s[256];
    const int tid = threadIdx.x;
    const int i = blockIdx.x * 256 + tid;
    const float m = mu[i], l = lv[i];
    z[i] = fmaf(expf(0.5f * l), eps[i], m);
    s[tid] = 1.0f + l - m * m - expf(l);
    __syncthreads();
    for (int off = 128; off > 0; off >>= 1) {
        if (tid < off) s[tid] += s[tid + off];
        __syncthreads();
    }
    if (tid == 0) part[blockIdx.x] = s[0];
}

// ---------------------------------------------------------------------------
// Kernel 5: G = IA = I - adjA*(1-I); also reset the grid-barrier state.
// ---------------------------------------------------------------------------
__global__ void sigrn_build_ia(const float* __restrict__ adjA,
                               float* __restrict__ G,
                               unsigned* __restrict__ bar) {
    const int i = blockIdx.x * 256 + threadIdx.x;
    if (i == 0) { bar[0] = 0u; bar[1] = 0u; }
    if (i < NG * NG) {
        const int g = i / NG, h = i % NG;
        G[i] = (g == h) ? 1.0f : -adjA[i];
    }
}

// ---------------------------------------------------------------------------
// Kernel 6: in-place Gauss-Jordan inverse (no pivoting: IA is diagonally
// dominant).  Persistent kernel, 32 blocks x 256, software grid barrier.
// The 4MB matrix stays resident in MI455X's 192MB L2.
// ---------------------------------------------------------------------------
__device__ __forceinline__ void grid_sync(unsigned* bar, unsigned nblk) {
    __syncthreads();
    if (threadIdx.x == 0) {
        __threadfence();
        const unsigned gen = atomicAdd(bar + 1, 0u);        // read current gen
        const unsigned prev = atomicAdd(bar, 1u);           // arrive
        if (prev == nblk - 1u) {
            atomicExch(bar, 0u);                            // reset count
            __threadfence();
            atomicAdd(bar + 1, 1u);                         // release
        } else {
            while (atomicAdd(bar + 1, 0u) == gen) {
                __builtin_amdgcn_s_sleep(2);
            }
        }
        __threadfence();
    }
    __syncthreads();
}

__global__ void sigrn_gj_inverse(float* __restrict__ G, unsigned* bar) {
    const int NBLK = 32;
    const int tid   = threadIdx.x;
    const int gtid  = blockIdx.x * 256 + tid;
    const int NT    = NBLK * 256;        // 8192 threads
    const int gwave = gtid >> 5;
    const int lane  = gtid & 31;
    const int NW    = NT >> 5;           // 256 waves

    for (int k = 0; k < NG; ++k) {
        // phase A: invert the pivot (single thread)
        if (gtid == 0) G[k * NG + k] = 1.0f / G[k * NG + k];
        grid_sync(bar, NBLK);
        // phase B: scale pivot row (j != k) by p = G[k][k]
        {
            const float p = G[k * NG + k];
            for (int j = gtid; j < NG; j += NT)
                if (j != k) G[k * NG + j] *= p;
        }
        grid_sync(bar, NBLK);
        // phase C: eliminate all other rows; G[i][k] <- -f*p
        for (int i = gwave; i < NG; i += NW) {
            if (i == k) continue;
            const float f = G[i * NG + k];
            for (int j = lane; j < NG; j += 32) {
                const float gk  = G[k * NG + j];     // == p at j==k
                const float gij = G[i * NG + j];
                G[i * NG + j] = (j == k) ? (-f * gk) : (gij - f * gk);
            }
        }
        grid_sync(bar, NBLK);
    }
}

// ---------------------------------------------------------------------------
// Kernel 7: per-block partial sums of (xn - x_rec)^2
// ---------------------------------------------------------------------------
__global__ void sigrn_rec_loss(const float* __restrict__ xn,
                               const float* __restrict__ xrec,
                               float* __restrict__ part) {
    __shared__ float s[256];
    const int tid = threadIdx.x;
    const int i = blockIdx.x * 256 + tid;
    const float d = xn[i] - xrec[i];
    s[tid] = d * d;
    __syncthreads();
    for (int off = 128; off > 0; off >>= 1) {
        if (tid < off) s[tid] += s[tid + off];
        __syncthreads();
    }
    if (tid == 0) part[blockIdx.x] = s[0];
}

// ---------------------------------------------------------------------------
// Kernel 8: final reduction of both losses
// ---------------------------------------------------------------------------
__global__ void sigrn_final(const float* __restrict__ partKL,
                            const float* __restrict__ partREC,
                            float* __restrict__ out) {
    __shared__ float sk[256], sr[256];
    const int tid = threadIdx.x;
    float a = 0.0f, b = 0.0f;
    for (int i = tid; i < BN / 256; i += 256) { a += partKL[i]; b += partREC[i]; }
    sk[tid] = a; sr[tid] = b;
    __syncthreads();
    for (int off = 128; off > 0; off >>= 1) {
        if (tid < off) { sk[tid] += sk[tid + off]; sr[tid] += sr[tid + off]; }
        __syncthreads();
    }
    if (tid == 0) {
        out[0] = sr[0] / (float)BN;             // loss_rec
        out[1] = -0.5f * sk[0] / (float)BN;     // loss_kl
    }
}

// ---------------------------------------------------------------------------
extern "C" void kernel_launch(void* const* d_in, const int* in_sizes, int n_in,
                              void* d_out, int out_size, void* d_ws, size_t ws_size,
                              hipStream_t stream) {
    const float* x     = (const float*)d_in[0];
    const float* gmean = (const float*)d_in[1];
    const float* gstd  = (const float*)d_in[2];
    const float* da    = (const float*)d_in[3];
    const float* eps   = (const float*)d_in[4];
    const float* adjA  = (const float*)d_in[5];
    const float* iw1   = (const float*)d_in[6];
    const float* ib1   = (const float*)d_in[7];
    const float* iw2   = (const float*)d_in[8];
    const float* ib2   = (const float*)d_in[9];
    const float* iw3   = (const float*)d_in[10];
    const float* ib3   = (const float*)d_in[11];
    const float* gw1   = (const float*)d_in[12];
    const float* gb1   = (const float*)d_in[13];
    const float* gw2   = (const float*)d_in[14];
    const float* gb2   = (const float*)d_in[15];
    const float* gw3   = (const float*)d_in[16];
    const float* gb3   = (const float*)d_in[17];

    float* out = (float*)d_out;
    float* o_xrec = out + 2;
    float* o_z    = out + 2 + (size_t)BN;
    float* o_mu   = out + 2 + (size_t)2 * BN;
    float* o_lv   = out + 2 + (size_t)3 * BN;

    // workspace layout (floats): xn | zp0 | zp1 | G(1e6) | partKL | partREC | bar
    float* wsf     = (float*)d_ws;
    float* xn      = wsf;
    float* zp0     = wsf + (size_t)BN;
    float* zp1     = wsf + (size_t)2 * BN;
    float* G       = wsf + (size_t)3 * BN;
    float* partKL  = G + (size_t)NG * NG;
    float* partREC = partKL + BN / 256;
    unsigned* bar  = (unsigned*)(partREC + BN / 256);
    float* zinv    = zp0;   // reuse: zp0 dead after the mu-GEMM

    sigrn_pre<<<BN / 256, 256, 0, stream>>>(x, gmean, gstd, da, xn);

    sigrn_mlp<<<BN / 64, 128, 0, stream>>>(xn, iw1, ib1, iw2, ib2, iw3, ib3,
                                           2, zp0, zp1);

    sigrn_gemm<<<(64 * 63) / 4, 128, 0, stream>>>(zp0, adjA, o_mu, 0);
    sigrn_gemm<<<(64 * 63) / 4, 128, 0, stream>>>(zp1, adjA, o_lv, 0);

    sigrn_z_kl<<<BN / 256, 256, 0, stream>>>(o_mu, o_lv, eps, o_z, partKL);

    sigrn_build_ia<<<(NG * NG + 255) / 256, 256, 0, stream>>>(adjA, G, bar);
    sigrn_gj_inverse<<<32, 256, 0, stream>>>(G, bar);

    sigrn_gemm<<<(64 * 63) / 4, 128, 0, stream>>>(o_z, G, zinv, 1);

    sigrn_mlp<<<BN / 64, 128, 0, stream>>>(zinv, gw1, gb1, gw2, gb2, gw3, gb3,
                                           1, o_xrec, nullptr);

    sigrn_rec_loss<<<BN / 256, 256, 0, stream>>>(xn, o_xrec, partREC);
    sigrn_final<<<1, 256, 0, stream>>>(partKL, partREC, out);
}